// VolumeRenderer_20418274525647
// MI455X (gfx1250) — compile-verified
//
#include <hip/hip_runtime.h>
#include <stdint.h>

// Volume rendering: B=8, R=16384, Z=128, C=3. Memory-bound (~322MB -> ~14us @ 23.3TB/s).
// Strategy: block = 256 rays; Z in 8 chunks of 16, double-buffered in LDS, staged with
// gfx1250 async global->LDS b128 copies (ASYNCcnt pipeline), compute is a per-ray scan.
// Inputs are streamed read-once -> NT temporal hint so the 256MB stream doesn't thrash
// the 192MB L2; weights (write-once) stored non-temporally.

#define ZS   128
#define CH   16
#define NCH  (ZS / CH)      // 8
#define RPB  256            // rays (threads) per block
#define SIG_STRIDE 20       // floats per ray row in LDS sigma buffer (16 data + 4 pad)
#define RAD_STRIDE 52       // floats per ray row in LDS radiance buffer (48 data + 4 pad)

// LDS byte offsets (flat layout)
#define OFF_SIG0  0
#define OFF_SIG1  (RPB * SIG_STRIDE * 4)
#define OFF_RAD0  (2 * RPB * SIG_STRIDE * 4)
#define OFF_RAD1  (OFF_RAD0 + RPB * RAD_STRIDE * 4)
#define OFF_ZV    (OFF_RAD0 + 2 * RPB * RAD_STRIDE * 4)
#define OFF_DIST  (OFF_ZV + ZS * 4)
#define LDS_BYTES (OFF_DIST + ZS * 4)   // 148480 B, 2 blocks/WGP fit in 320KB

typedef float v4f __attribute__((ext_vector_type(4)));   // true vector type for NT store

// ---- gfx1250 async global->LDS path (progressive enhancement) ----------------
#if defined(__has_builtin)
#  if __has_builtin(__builtin_amdgcn_global_load_async_to_lds_b128) && \
      __has_builtin(__builtin_amdgcn_s_wait_asynccnt)
#    define HAVE_ASYNC 1
#  endif
#endif
#ifndef HAVE_ASYNC
#  define HAVE_ASYNC 0
#endif

#if HAVE_ASYNC
typedef int v4i __attribute__((ext_vector_type(4)));
typedef __attribute__((address_space(1))) v4i GV4;   // global int4
typedef __attribute__((address_space(3))) v4i LV4;   // LDS int4
// cpol = 1 -> TH_LOAD_NT (gfx12 CPol: TH[2:0], scope[4:3]); inputs are read-once.
#define ASYNC_CP16(lp, gp) \
  __builtin_amdgcn_global_load_async_to_lds_b128((GV4*)(gp), (LV4*)(lp), 0, 1)
#define WAIT_ASYNC(n) __builtin_amdgcn_s_wait_asynccnt(n)
#else
// synchronous fallback: plain b128 load + ds store
#define ASYNC_CP16(lp, gp) (*(v4f*)(lp) = *(const v4f*)(gp))
#define WAIT_ASYNC(n) ((void)0)
#endif
// -----------------------------------------------------------------------------

// Stage one Z-chunk (16 samples) of sigma + radiance for the block's 256 rays.
// Linear 16B-piece index k4 = i*256 + t gives contiguous global runs per ray
// (64B sigma rows, 192B radiance rows) -> coalesced, full-cacheline traffic.
__device__ __forceinline__ void issue_chunk(char* smem, int buf, int c, int t,
                                            const char* sig_g, const char* rad_g) {
  char* sigL = smem + (buf ? OFF_SIG1 : OFF_SIG0);
  char* radL = smem + (buf ? OFF_RAD1 : OFF_RAD0);
#pragma unroll
  for (int i = 0; i < 4; ++i) {          // sigma: 256 rays * 64B = 1024 pieces
    int k4  = i * RPB + t;
    int ray = k4 >> 2, q = k4 & 3;
    ASYNC_CP16(sigL + ray * (SIG_STRIDE * 4) + q * 16,
               sig_g + (size_t)ray * (ZS * 4) + c * (CH * 4) + q * 16);
  }
#pragma unroll
  for (int i = 0; i < 12; ++i) {         // radiance: 256 rays * 192B = 3072 pieces
    int k4  = i * RPB + t;
    int ray = k4 / 12, j = k4 - ray * 12;
    ASYNC_CP16(radL + ray * (RAD_STRIDE * 4) + j * 16,
               rad_g + (size_t)ray * (ZS * 3 * 4) + c * (CH * 3 * 4) + j * 16);
  }
}

__global__ __launch_bounds__(RPB)
void VolumeRenderer_volrender(const float* __restrict__ sigmas,
                              const float* __restrict__ radiances,
                              const float* __restrict__ zvals,
                              float* __restrict__ out, int nrays) {
  __shared__ char smem[LDS_BYTES];
  const int t    = threadIdx.x;
  const int ray0 = blockIdx.x * RPB;
  const int g    = ray0 + t;

  const char* sig_g = (const char*)(sigmas    + (size_t)ray0 * ZS);
  const char* rad_g = (const char*)(radiances + (size_t)ray0 * ZS * 3);
  float* zvL = (float*)(smem + OFF_ZV);
  float* dsL = (float*)(smem + OFF_DIST);

  // Prologue: start chunk 0 fetch, stage z_vals, derive dists (last interval repeated).
  issue_chunk(smem, 0, 0, t, sig_g, rad_g);
  if (t < ZS) zvL[t] = zvals[t];
  __syncthreads();
  if (t < ZS)
    dsL[t] = (t < ZS - 1) ? (zvL[t + 1] - zvL[t]) : (zvL[ZS - 1] - zvL[ZS - 2]);
  // dsL visibility is covered by the barrier inside the first loop iteration.

  float T = 1.0f, accR = 0.f, accG = 0.f, accB = 0.f, accD = 0.f;
  float* wout = out + (size_t)nrays * 4 + (size_t)g * ZS;   // weights region

  for (int c = 0; c < NCH; ++c) {
    const int buf = c & 1;
    if (c + 1 < NCH) {                   // prefetch next chunk into other buffer
      issue_chunk(smem, buf ^ 1, c + 1, t, sig_g, rad_g);
      WAIT_ASYNC(16);                    // chunk c complete, chunk c+1 in flight
    } else {
      WAIT_ASYNC(0);
    }
    __syncthreads();                     // all waves' chunk-c data visible in LDS

    const v4f* s4 = (const v4f*)(smem + (buf ? OFF_SIG1 : OFF_SIG0) +
                                 t * (SIG_STRIDE * 4));
    const v4f* r4 = (const v4f*)(smem + (buf ? OFF_RAD1 : OFF_RAD0) +
                                 t * (RAD_STRIDE * 4));
#pragma unroll
    for (int i = 0; i < 4; ++i) {        // 4 samples per group
      v4f sv4 = s4[i];
      v4f ra = r4[3 * i], rb = r4[3 * i + 1], rc = r4[3 * i + 2];
      float sv[4]  = {sv4.x, sv4.y, sv4.z, sv4.w};
      float rr[12] = {ra.x, ra.y, ra.z, ra.w,
                      rb.x, rb.y, rb.z, rb.w,
                      rc.x, rc.y, rc.z, rc.w};
      float wv[4];
      const int zb = c * CH + i * 4;
#pragma unroll
      for (int u = 0; u < 4; ++u) {
        float s = fmaxf(sv[u], 0.0f);                 // relu(sigma)
        float e = __expf(-s * dsL[zb + u]);           // 1-alpha (pre-eps)
        float w = (1.0f - e) * T;                     // alpha * T
        wv[u] = w;
        accR += w * rr[3 * u + 0];
        accG += w * rr[3 * u + 1];
        accB += w * rr[3 * u + 2];
        accD += w * zvL[zb + u];
        T *= (e + 1e-10f);                            // cumprod(1-alpha+eps)
      }
      v4f w4 = {wv[0], wv[1], wv[2], wv[3]};
      __builtin_nontemporal_store(w4, (v4f*)(wout + zb));  // write-once stream
    }
    __syncthreads();                     // buffer safe to overwrite next iter
  }

  out[(size_t)g * 3 + 0] = accR;                      // rgb  [B,R,3]
  out[(size_t)g * 3 + 1] = accG;
  out[(size_t)g * 3 + 2] = accB;
  out[(size_t)nrays * 3 + g] = accD;                  // depth [B,R,1]
}

extern "C" void kernel_launch(void* const* d_in, const int* in_sizes, int n_in,
                              void* d_out, int out_size, void* d_ws, size_t ws_size,
                              hipStream_t stream) {
  (void)n_in; (void)out_size; (void)d_ws; (void)ws_size;
  const float* sigmas    = (const float*)d_in[0];
  const float* radiances = (const float*)d_in[1];
  const float* zvals     = (const float*)d_in[2];
  float* out = (float*)d_out;
  const int nrays  = in_sizes[0] / ZS;   // B*R = 131072
  const int blocks = nrays / RPB;        // 512 (exact for B=8, R=16384)
  hipLaunchKernelGGL(VolumeRenderer_volrender, dim3(blocks), dim3(RPB), 0, stream,
                     sigmas, radiances, zvals, out, nrays);
}